// HetFlow_62405874811918
// MI455X (gfx1250) — compile-verified
//
#include <hip/hip_runtime.h>
#include <hip/hip_bf16.h>

// ---------------- problem constants (match reference) ----------------
#define NB   16            // batch
#define NN   30000         // nodes
#define TT   6             // input time steps
#define EE   120000        // edges per etype
#define C1_  8
#define C2_  11
#define HID_ 128
#define S1_  64            // B * 4  (block1 slices)
#define S2_  32            // B * 2  (block2 slices)
#define F1_  (S1_ * C1_)   // 512 floats per node row, block1
#define F2_  (S2_ * C2_)   // 352 floats per node row, block2

typedef float v2f __attribute__((ext_vector_type(2)));
typedef float v8f __attribute__((ext_vector_type(8)));

static inline long divup_l(long a, long b) { return (a + b - 1) / b; }

// ---------------- temporal conv 1 (GLU, 1->8 ch) -> Xn1 [N, 64, 8] ----------------
__global__ void tconv1_kernel(const float* __restrict__ x,   // [B,N,T,1]
                              const float* __restrict__ w,   // [16,1,3,1]
                              const float* __restrict__ bia, // [16]
                              float* __restrict__ Xn1)
{
    long idx = (long)blockIdx.x * blockDim.x + threadIdx.x;   // over N*64
    if (idx >= (long)NN * S1_) return;
    int n = (int)(idx >> 6);
    int s = (int)(idx & 63);
    int b = s >> 2, t = s & 3;
    const float* xp = x + ((long)b * NN + n) * TT;
    float x0 = xp[t], x1 = xp[t + 1], x2 = xp[t + 2];
    float* dst = Xn1 + (long)n * F1_ + s * C1_;
    #pragma unroll
    for (int c = 0; c < C1_; ++c) {
        float p = bia[c]       + w[c*3]        * x0 + w[c*3+1]        * x1 + w[c*3+2]        * x2;
        float q = bia[c + C1_] + w[(c+C1_)*3]  * x0 + w[(c+C1_)*3+1]  * x1 + w[(c+C1_)*3+2]  * x2;
        float al = (c == 0) ? x2 : 0.f;   // pad-align: channel 0 = x at t+KT-1
        dst[c] = (p + al) * (1.f / (1.f + __expf(-q)));
    }
}

// ---------------- degree / norm ----------------
__global__ void degree_kernel(const int* __restrict__ src, const int* __restrict__ dst,
                              float* __restrict__ degout, float* __restrict__ degin)
{
    int e = blockIdx.x * blockDim.x + threadIdx.x;
    if (e >= EE) return;
    atomicAdd(&degout[src[e]], 1.f);
    atomicAdd(&degin[dst[e]], 1.f);
}

__global__ void norm_kernel(const int* __restrict__ src, const int* __restrict__ dst,
                            const float* __restrict__ degout, const float* __restrict__ degin,
                            float* __restrict__ nrm)
{
    int e = blockIdx.x * blockDim.x + threadIdx.x;
    if (e >= EE) return;
    nrm[e] = rsqrtf(fmaxf(degout[src[e]], 1.f)) * rsqrtf(fmaxf(degin[dst[e]], 1.f));
}

// ---------------- prop: Out[dst,:] += -norm * Tf[src,:] ----------------
// One wave per TWO edges: prefetch edge1's source row (global_prefetch_b8)
// while edge0's gather + global_atomic_add_f32 stream is in flight.
__global__ void prop_kernel(const float* __restrict__ Tf, float* __restrict__ Out,
                            const int* __restrict__ src, const int* __restrict__ dst,
                            const float* __restrict__ nrm, int F)
{
    int wave = blockIdx.x * (blockDim.x >> 5) + (threadIdx.x >> 5);
    if (wave >= EE / 2) return;
    int lane = threadIdx.x & 31;
    int e0 = wave * 2, e1 = e0 + 1;
    const float* a0 = Tf + (long)src[e0] * F;
    const float* a1 = Tf + (long)src[e1] * F;
    float* o0 = Out + (long)dst[e0] * F;
    float* o1 = Out + (long)dst[e1] * F;
    float nm0 = -nrm[e0], nm1 = -nrm[e1];
    // lane-strided prefetch covers the whole second row (F/32 floats per lane)
    __builtin_prefetch(a1 + lane * (F >> 5), 0, 0);
    for (int i = lane; i < F; i += 32)
        atomicAdd(&o0[i], nm0 * a0[i]);
    for (int i = lane; i < F; i += 32)
        atomicAdd(&o1[i], nm1 * a1[i]);
}

// ---------------- T2 = 2*T2 - Xn ----------------
__global__ void cheb_fin_kernel(float* __restrict__ T2, const float* __restrict__ Xn, long n)
{
    long i = (long)blockIdx.x * blockDim.x + threadIdx.x;
    if (i < n) T2[i] = 2.f * T2[i] - Xn[i];
}

// ---------------- init out with bias[0]+bias[1] ----------------
__global__ void init_bias2_kernel(float* __restrict__ out, const float* __restrict__ bia,
                                  long total, int cout)
{
    long i = (long)blockIdx.x * blockDim.x + threadIdx.x;
    if (i >= total) return;
    int co = (int)(i % cout);
    out[i] = bia[co] + bia[cout + co];
}

__global__ void init_vec_kernel(float* __restrict__ out, const float* __restrict__ v,
                                long total, int h)
{
    long i = (long)blockIdx.x * blockDim.x + threadIdx.x;
    if (i >= total) return;
    out[i] = v[i % h];
}

// ---------------- specialized f32 WMMA GEMM accumulate: C += A(MxK) @ B(KxN) ----------------
// One wave per 16x16 C tile; V_WMMA_F32_16X16X4_F32 chained over K in steps of 4.
// A row stride == K (true at every call site). M is a multiple of 16.
// Ragged K / NCOLS handled branch-free: clamped (in-bounds) loads + lane selects,
// so EXEC stays all-1s through the WMMA chain with no saveexec churn.
template<int K, int NCOLS>
__global__ void wmma_gemm_acc_t(const float* __restrict__ A,
                                const float* __restrict__ B, int ldb,
                                float* __restrict__ C, int ldc, long Mtiles)
{
    long wave = (long)blockIdx.x * (blockDim.x >> 5) + (threadIdx.x >> 5);
    if (wave >= Mtiles) return;                 // wave-uniform exit
    const int lane = threadIdx.x & 31;
    const int hi   = lane >> 4;                 // 0: K pair {k0,k0+1}; 1: {k0+2,k0+3}
    const int l16  = lane & 15;
    const long mbase = wave * 16;
    const int  nbase = blockIdx.y * 16;
    const int  ccol  = nbase + l16;
    constexpr bool FULLC = (NCOLS % 16 == 0);
    const bool colok = FULLC ? true : (ccol < NCOLS);
    const int  csafe = FULLC ? ccol : (ccol < NCOLS ? ccol : NCOLS - 1);

    const long crow0 = mbase + hi * 8;          // C/D: VGPR j -> M = j + 8*hi, N = l16
    v8f acc;
    #pragma unroll
    for (int j = 0; j < 8; ++j) {
        float cv = C[(crow0 + j) * (long)ldc + csafe];   // always in-bounds
        acc[j] = colok ? cv : 0.f;
    }

    const float* arow = A + (mbase + l16) * (long)K;
    #pragma unroll
    for (int k0 = 0; k0 < K; k0 += 4) {
        const int ka = k0 + 2 * hi;
        v2f a, b;
        if constexpr (K % 4 == 0) {
            a = *(const v2f*)(arow + ka);        // 8B-aligned: K even, ka even
            float bx = B[(long)ka * ldb + csafe];
            float by = B[(long)(ka + 1) * ldb + csafe];
            b.x = colok ? bx : 0.f;
            b.y = colok ? by : 0.f;
        } else {
            const bool k0ok = (ka < K), k1ok = (ka + 1 < K);
            const int ka0 = k0ok ? ka : K - 1;
            const int ka1 = k1ok ? ka + 1 : K - 1;
            float ax = arow[ka0], ay = arow[ka1];
            a.x = k0ok ? ax : 0.f;
            a.y = k1ok ? ay : 0.f;
            float bx = B[(long)ka0 * ldb + csafe];
            float by = B[(long)ka1 * ldb + csafe];
            b.x = (k0ok && colok) ? bx : 0.f;
            b.y = (k1ok && colok) ? by : 0.f;
        }
        acc = __builtin_amdgcn_wmma_f32_16x16x4_f32(false, a, false, b, (short)0, acc,
                                                    false, false);
    }

    if constexpr (FULLC) {
        #pragma unroll
        for (int j = 0; j < 8; ++j)
            C[(crow0 + j) * (long)ldc + ccol] = acc[j];
    } else {
        if (colok) {
            #pragma unroll
            for (int j = 0; j < 8; ++j)
                C[(crow0 + j) * (long)ldc + ccol] = acc[j];
        }
    }
}

// ---------------- temporal conv 2 (align 1x1 16->11, GLU 16->11) -> Xn2 [N, 32, 11] ----------------
__global__ void tconv2_kernel(const float* __restrict__ outS, const float* __restrict__ outT,
                              const float* __restrict__ w,  const float* __restrict__ wb,
                              const float* __restrict__ aw, const float* __restrict__ ab,
                              float* __restrict__ Xn2)
{
    long idx = (long)blockIdx.x * blockDim.x + threadIdx.x;   // over N*32
    if (idx >= (long)NN * S2_) return;
    int n  = (int)(idx >> 5);
    int s2 = (int)(idx & 31);
    int b = s2 >> 1, t2 = s2 & 1;

    float xin[16][3];
    #pragma unroll
    for (int k = 0; k < 3; ++k) {
        int s1 = b * 4 + (t2 + k);
        const float* ps = outS + (long)n * F1_ + s1 * C1_;
        const float* pt = outT + (long)n * F1_ + s1 * C1_;
        #pragma unroll
        for (int c = 0; c < 8; ++c) { xin[c][k] = ps[c]; xin[c + 8][k] = pt[c]; }
    }

    float* dstp = Xn2 + (long)n * F2_ + s2 * C2_;
    for (int co = 0; co < C2_; ++co) {
        float p = wb[co], q = wb[co + C2_];
        #pragma unroll
        for (int c = 0; c < 16; ++c) {
            #pragma unroll
            for (int k = 0; k < 3; ++k) {
                p += w[(co * 16 + c) * 3 + k]          * xin[c][k];
                q += w[((co + C2_) * 16 + c) * 3 + k]  * xin[c][k];
            }
        }
        float al = ab[co];
        #pragma unroll
        for (int c = 0; c < 16; ++c) al += aw[co * 16 + c] * xin[c][2];
        dstp[co] = (p + al) * (1.f / (1.f + __expf(-q)));
    }
}

// ---------------- pack Z [B*N, 44]: j = c*2 + t; c<11 from out2S else out2T ----------------
__global__ void pack_z_kernel(const float* __restrict__ o2S, const float* __restrict__ o2T,
                              float* __restrict__ Z)
{
    long i = (long)blockIdx.x * blockDim.x + threadIdx.x;   // over B*N*44
    if (i >= (long)NB * NN * 44) return;
    int j = (int)(i % 44);
    long r = i / 44;
    int n = (int)(r % NN);
    int b = (int)(r / NN);
    int t = j & 1, c = j >> 1;
    float v = (c < C2_) ? o2S[(long)n * F2_ + (b * 2 + t) * C2_ + c]
                        : o2T[(long)n * F2_ + (b * 2 + t) * C2_ + (c - C2_)];
    Z[i] = v;
}

// ---------------- host driver ----------------
template<int K, int NCOLS>
static void run_gemm(const float* A, const float* B, int ldb,
                     float* C, int ldc, long rows, hipStream_t stream)
{
    long mtiles = rows / 16;                   // rows always multiple of 16 here
    dim3 grid((unsigned)divup_l(mtiles, 4), (unsigned)((NCOLS + 15) / 16));
    wmma_gemm_acc_t<K, NCOLS><<<grid, 128, 0, stream>>>(A, B, ldb, C, ldc, mtiles);
}

template<int CIN>
static void run_hetcheb(const float* Xn, int S,
                        const int* edges,        // [2,2,E]
                        const float* W,          // [2,3,CIN,CIN]
                        const float* bias,       // [2,CIN]
                        float* out,              // [N,S,CIN] accumulate target
                        float* T1, float* T2, float* deg, float* nrm,
                        hipStream_t stream)
{
    const int F = S * CIN;
    const long rows = (long)NN * S;
    const long tot  = rows * CIN;
    init_bias2_kernel<<<(unsigned)divup_l(tot, 256), 256, 0, stream>>>(out, bias, tot, CIN);

    for (int e = 0; e < 2; ++e) {
        const int* src = edges + (long)(e * 2 + 0) * EE;
        const int* dst = edges + (long)(e * 2 + 1) * EE;
        hipMemsetAsync(deg, 0, 2 * NN * sizeof(float), stream);
        degree_kernel<<<(EE + 255) / 256, 256, 0, stream>>>(src, dst, deg, deg + NN);
        norm_kernel<<<(EE + 255) / 256, 256, 0, stream>>>(src, dst, deg, deg + NN, nrm);

        const int prop_blocks = (EE / 2 * 32 + 255) / 256;   // two edges per wave
        hipMemsetAsync(T1, 0, (size_t)NN * F * sizeof(float), stream);
        prop_kernel<<<prop_blocks, 256, 0, stream>>>(Xn, T1, src, dst, nrm, F);
        hipMemsetAsync(T2, 0, (size_t)NN * F * sizeof(float), stream);
        prop_kernel<<<prop_blocks, 256, 0, stream>>>(T1, T2, src, dst, nrm, F);
        cheb_fin_kernel<<<(unsigned)divup_l((long)NN * F, 256), 256, 0, stream>>>(T2, Xn, (long)NN * F);

        const float* We = W + (long)e * 3 * CIN * CIN;
        run_gemm<CIN, CIN>(Xn, We,                 CIN, out, CIN, rows, stream);
        run_gemm<CIN, CIN>(T1, We + CIN * CIN,     CIN, out, CIN, rows, stream);
        run_gemm<CIN, CIN>(T2, We + 2 * CIN * CIN, CIN, out, CIN, rows, stream);
    }
}

extern "C" void kernel_launch(void* const* d_in, const int* in_sizes, int n_in,
                              void* d_out, int out_size, void* d_ws, size_t ws_size,
                              hipStream_t stream)
{
    const float* x        = (const float*)d_in[0];
    const int*   hg_edges = (const int*)  d_in[1];
    const int*   tg_edges = (const int*)  d_in[2];
    const float* tconv1_w = (const float*)d_in[3];
    const float* tconv1_b = (const float*)d_in[4];
    const float* h1_W     = (const float*)d_in[5];
    const float* h1_b     = (const float*)d_in[6];
    const float* t1_W     = (const float*)d_in[7];
    const float* t1_b     = (const float*)d_in[8];
    const float* align2_w = (const float*)d_in[9];
    const float* align2_b = (const float*)d_in[10];
    const float* tconv2_w = (const float*)d_in[11];
    const float* tconv2_b = (const float*)d_in[12];
    const float* h2_W     = (const float*)d_in[13];
    const float* h2_b     = (const float*)d_in[14];
    const float* t2_W     = (const float*)d_in[15];
    const float* t2_b     = (const float*)d_in[16];
    const float* out_w    = (const float*)d_in[17];
    const float* out_b    = (const float*)d_in[18];
    float* out = (float*)d_out;

    // workspace layout (floats), lifetime-based reuse
    const long SZ1 = (long)NN * F1_;   // 15.36M floats
    const long SZ2 = (long)NN * F2_;   // 10.56M floats
    float* ws   = (float*)d_ws;
    float* Xn1  = ws;                  // block1 features; later overlaid by Z
    float* T1   = ws + SZ1;            // prop buffer (both blocks); head overlaid by Z at the end
    float* T2   = ws + 2 * SZ1;        // prop buffer (both blocks)
    float* o1S  = ws + 3 * SZ1;        // block1 hg out; later block2 hg out
    float* o1T  = ws + 4 * SZ1;        // block1 tg out; later block2 tg out
    float* Xn2  = ws + 5 * SZ1;
    float* deg  = Xn2 + SZ2;           // 2*N floats
    float* nrm  = deg + 2 * NN;        // E floats
    float* Z    = ws;                  // [B*N,44] = 21.12M floats, fits in Xn1+part of T1 (freed)

    // ---- block 1 ----
    tconv1_kernel<<<(unsigned)divup_l((long)NN * S1_, 256), 256, 0, stream>>>(
        x, tconv1_w, tconv1_b, Xn1);
    run_hetcheb<C1_>(Xn1, S1_, hg_edges, h1_W, h1_b, o1S, T1, T2, deg, nrm, stream);
    run_hetcheb<C1_>(Xn1, S1_, tg_edges, t1_W, t1_b, o1T, T1, T2, deg, nrm, stream);

    // ---- block 2 ----
    tconv2_kernel<<<(unsigned)divup_l((long)NN * S2_, 256), 256, 0, stream>>>(
        o1S, o1T, tconv2_w, tconv2_b, align2_w, align2_b, Xn2);
    run_hetcheb<C2_>(Xn2, S2_, hg_edges, h2_W, h2_b, o1S /*reuse as o2S*/, T1, T2, deg, nrm, stream);
    run_hetcheb<C2_>(Xn2, S2_, tg_edges, t2_W, t2_b, o1T /*reuse as o2T*/, T1, T2, deg, nrm, stream);

    // ---- output linear: [B*N,44] @ [44,128] + out_b ----
    pack_z_kernel<<<(unsigned)divup_l((long)NB * NN * 44, 256), 256, 0, stream>>>(o1S, o1T, Z);
    const long orows = (long)NB * NN;
    init_vec_kernel<<<(unsigned)divup_l(orows * HID_, 256), 256, 0, stream>>>(out, out_b, orows * HID_, HID_);
    run_gemm<44, HID_>(Z, out_w, HID_, out, HID_, orows, stream);
}